// Memory_Shift_56831007260832
// MI455X (gfx1250) — compile-verified
//
#include <hip/hip_runtime.h>

#define L_ 2
#define B_ 8
#define T_ 256
#define N_ 512
#define K_ 16
#define M_ 10000
#define H_ 256
#define R_ 500
#define NN 64            // n-rows per scan workgroup
#define CH (N_ / NN)     // 8 chunks per batch

typedef __attribute__((ext_vector_type(2))) float v2f;
typedef __attribute__((ext_vector_type(8))) float v8f;

// ---------------------------------------------------------------------------
// Kernel 1: gather+sum heads/tails embeddings into cat[b,n,0:2H]; gather rel.
// One block per (b,n), 256 threads (= H lanes).
// ---------------------------------------------------------------------------
__global__ void k_gather(const float* __restrict__ se,
                         const float* __restrict__ rel_table,
                         const int* __restrict__ heads,
                         const int* __restrict__ tails,
                         const int* __restrict__ relidx,
                         float* __restrict__ cat,
                         float* __restrict__ rel) {
  const int bn = blockIdx.x;            // b*N + n
  const int h  = threadIdx.x;           // 0..H-1
  const int b  = bn / N_;
  const float* seb = se + (size_t)b * M_ * H_;
  const int* hp = heads + (size_t)bn * K_;
  const int* tp = tails + (size_t)bn * K_;
  float ah = 0.f, at = 0.f;
#pragma unroll
  for (int k = 0; k < K_; ++k) {
    ah += seb[(size_t)hp[k] * H_ + h];
    at += seb[(size_t)tp[k] * H_ + h];
  }
  cat[(size_t)bn * (2 * H_) + h]      = ah;
  cat[(size_t)bn * (2 * H_) + H_ + h] = at;
  rel[(size_t)bn * H_ + h] = rel_table[(size_t)relidx[bn] * H_ + h];
}

// ---------------------------------------------------------------------------
// Kernel 2: C[Mtot x 256] = (A[Mtot x Kt] @ W[256 x Kt]^T + bias) * scale,
// optionally zeroed per-row by rowmask.  One wave (32 lanes) per 16x16 tile,
// fp32 WMMA 16x16x4.  grid = (Mtot/16, 256/16), block = 32.
// ---------------------------------------------------------------------------
__global__ void k_gemm_nt(const float* __restrict__ A,
                          const float* __restrict__ W,
                          const float* __restrict__ bias,
                          const int* __restrict__ rowmask,
                          float* __restrict__ C,
                          int Kt, float scale) {
  const int lane = threadIdx.x;
  const int half = lane >> 4;      // K-half selector (lanes 16-31 hold K=2,3)
  const int l15  = lane & 15;
  const int row0 = blockIdx.x * 16;
  const int col0 = blockIdx.y * 16;
  const float* arow = A + (size_t)(row0 + l15) * Kt + half * 2;
  const float* wrow = W + (size_t)(col0 + l15) * Kt + half * 2;
  v8f acc = {};
#pragma unroll 4
  for (int k0 = 0; k0 < Kt * 1; k0 += 4) {
    v2f a; a.x = arow[k0]; a.y = arow[k0 + 1];
    v2f bm; bm.x = wrow[k0]; bm.y = wrow[k0 + 1];
    acc = __builtin_amdgcn_wmma_f32_16x16x4_f32(false, a, false, bm,
                                                (short)0, acc, false, false);
  }
  const int col = col0 + l15;
  const float bv = bias ? bias[col] : 0.f;
#pragma unroll
  for (int r = 0; r < 8; ++r) {
    const int row = row0 + r + half * 8;
    float v = (acc[r] + bv) * scale;
    if (rowmask && rowmask[row] == 0) v = 0.f;
    C[(size_t)row * H_ + col] = v;
  }
}

// ---------------------------------------------------------------------------
// Kernel 3: batched scores[b,t,n] = sum_h q[b,t,h]*k[b,n,h].
// grid = (T/16, N/16, B), block = 32 (one wave per 16x16 tile), WMMA f32.
// ---------------------------------------------------------------------------
__global__ void k_scores(const float* __restrict__ q,
                         const float* __restrict__ kk,
                         float* __restrict__ scores) {
  const int lane = threadIdx.x;
  const int half = lane >> 4;
  const int l15  = lane & 15;
  const int b  = blockIdx.z;
  const int t0 = blockIdx.x * 16;
  const int n0 = blockIdx.y * 16;
  const float* arow = q  + ((size_t)b * T_ + t0 + l15) * H_ + half * 2;
  const float* brow = kk + ((size_t)b * N_ + n0 + l15) * H_ + half * 2;
  v8f acc = {};
#pragma unroll 4
  for (int k0 = 0; k0 < H_; k0 += 4) {
    v2f a; a.x = arow[k0]; a.y = arow[k0 + 1];
    v2f bm; bm.x = brow[k0]; bm.y = brow[k0 + 1];
    acc = __builtin_amdgcn_wmma_f32_16x16x4_f32(false, a, false, bm,
                                                (short)0, acc, false, false);
  }
#pragma unroll
  for (int r = 0; r < 8; ++r)
    scores[((size_t)b * T_ + t0 + r + half * 8) * N_ + n0 + l15] = acc[r];
}

// ---------------------------------------------------------------------------
// Kernel 4: masked softmax over N=512, in place.  One block per (b,t), 256 thr.
// ---------------------------------------------------------------------------
__global__ void k_softmax(float* __restrict__ scores,
                          const int* __restrict__ tri_mask) {
  const int bt = blockIdx.x;
  const int b  = bt / T_;
  const int tid = threadIdx.x;
  float* row = scores + (size_t)bt * N_;
  const int* mrow = tri_mask + (size_t)b * N_;
  __shared__ float red[256];
  float v0 = (mrow[tid]       == 1) ? row[tid]       : -1.0e9f;
  float v1 = (mrow[tid + 256] == 1) ? row[tid + 256] : -1.0e9f;
  red[tid] = fmaxf(v0, v1);
  __syncthreads();
  for (int s = 128; s > 0; s >>= 1) {
    if (tid < s) red[tid] = fmaxf(red[tid], red[tid + s]);
    __syncthreads();
  }
  const float m = red[0];
  __syncthreads();
  const float e0 = __expf(v0 - m), e1 = __expf(v1 - m);
  red[tid] = e0 + e1;
  __syncthreads();
  for (int s = 128; s > 0; s >>= 1) {
    if (tid < s) red[tid] += red[tid + s];
    __syncthreads();
  }
  const float inv = 1.f / red[0];
  row[tid]       = e0 * inv;
  row[tid + 256] = e1 * inv;
}

// ---------------------------------------------------------------------------
// Kernel 5: the T=256 sequential scan.  One block per (b, n-chunk of NN=64),
// 256 threads; u tile lives in LDS (64 KB) for all timesteps.
// Thread `tid` owns column h=tid of the tile for the elementwise phase.
// Per step: per-row dot with wu_w (wave-parallel rows, lane-parallel H),
// gated update, and per-chunk partial output qtp[block][t][h].
// ---------------------------------------------------------------------------
__global__ void k_scan(const float* __restrict__ u0,
                       const float* __restrict__ wd,
                       const float* __restrict__ ug,
                       const float* __restrict__ pasi,
                       const float* __restrict__ h_last,
                       const float* __restrict__ aerfa_w,
                       const float* __restrict__ aerfa_b,
                       const float* __restrict__ wu_w,
                       const float* __restrict__ wu_b,
                       float* __restrict__ qtp) {
  const int b     = blockIdx.x / CH;
  const int chunk = blockIdx.x % CH;
  const int n0    = chunk * NN;
  const int tid   = threadIdx.x;

  __shared__ float u[NN * H_];   // 64 KB state tile
  __shared__ float wwu[H_];
  __shared__ float wun[NN];
  __shared__ float pas[NN];
  __shared__ float red[256];
  __shared__ float a_sh;

  for (int idx = tid; idx < NN * H_; idx += 256)
    u[idx] = u0[((size_t)b * N_ + n0) * H_ + idx];
  wwu[tid] = wu_w[tid];

  // a = sigmoid(h_last[b, T-1, :] . aerfa_w + aerfa_b)  (block reduction)
  red[tid] = h_last[((size_t)b * T_ + (T_ - 1)) * H_ + tid] * aerfa_w[tid];
  __syncthreads();
  for (int s = 128; s > 0; s >>= 1) {
    if (tid < s) red[tid] += red[tid + s];
    __syncthreads();
  }
  if (tid == 0) a_sh = 1.f / (1.f + __expf(-(red[0] + aerfa_b[0])));
  __syncthreads();

  const float a   = a_sh;
  const float wub = wu_b[0];
  const int wave = tid >> 5, lane = tid & 31;

  for (int t = 0; t < T_; ++t) {
    // wu[n] = u[n,:].wu_w + wu_b  — each wave handles NN/8 = 8 rows
#pragma unroll
    for (int rr = 0; rr < NN / 8; ++rr) {
      const int n = wave * (NN / 8) + rr;
      const float* ur = &u[n * H_];
      float p = 0.f;
#pragma unroll
      for (int i = 0; i < H_ / 32; ++i)
        p += ur[lane + i * 32] * wwu[lane + i * 32];
      for (int off = 16; off > 0; off >>= 1)
        p += __shfl_xor(p, off, 32);
      if (lane == 0) wun[n] = p + wub;
    }
    if (tid < NN)
      pas[tid] = pasi[((size_t)b * T_ + t) * N_ + n0 + tid];
    const float wdt = wd[((size_t)b * T_ + t) * H_ + tid];
    const float ugt = ug[((size_t)b * T_ + t) * H_ + tid];
    __syncthreads();

    float qt = 0.f;
#pragma unroll 4
    for (int n = 0; n < NN; ++n) {
      float un = u[n * H_ + tid];
      const float beta = a / (1.f + __expf(-(wdt + wun[n])));
      un = un * (1.f - beta) + ugt * beta;
      u[n * H_ + tid] = un;
      qt += un * pas[n];
    }
    qtp[((size_t)blockIdx.x * T_ + t) * H_ + tid] = qt;
    __syncthreads();
  }
}

// ---------------------------------------------------------------------------
// Kernel 6: reduce chunk partials -> out[b,t,h].
// ---------------------------------------------------------------------------
__global__ void k_reduce(const float* __restrict__ qtp,
                         float* __restrict__ out) {
  const int idx = blockIdx.x * 256 + threadIdx.x;   // over B*T*H
  const int b  = idx / (T_ * H_);
  const int th = idx % (T_ * H_);
  float s = 0.f;
#pragma unroll
  for (int c = 0; c < CH; ++c)
    s += qtp[(size_t)(b * CH + c) * (T_ * H_) + th];
  out[idx] = s;
}

// ---------------------------------------------------------------------------
extern "C" void kernel_launch(void* const* d_in, const int* in_sizes, int n_in,
                              void* d_out, int out_size, void* d_ws, size_t ws_size,
                              hipStream_t stream) {
  const float* hidden   = (const float*)d_in[0];
  const float* se       = (const float*)d_in[1];
  const float* rel_tab  = (const float*)d_in[2];
  const float* squeeze_w = (const float*)d_in[3];
  const float* squeeze_b = (const float*)d_in[4];
  const float* aerfa_w  = (const float*)d_in[5];
  const float* aerfa_b  = (const float*)d_in[6];
  const float* wq_w     = (const float*)d_in[7];
  const float* wq_b     = (const float*)d_in[8];
  const float* wk_w     = (const float*)d_in[9];
  const float* wk_b     = (const float*)d_in[10];
  const float* wd_w     = (const float*)d_in[11];
  const float* wd_b     = (const float*)d_in[12];
  const float* wu_w     = (const float*)d_in[13];
  const float* wu_b     = (const float*)d_in[14];
  const float* wg_w     = (const float*)d_in[15];
  const float* wg_b     = (const float*)d_in[16];
  const int*   heads    = (const int*)d_in[17];
  const int*   tails    = (const int*)d_in[18];
  const int*   tri_mask = (const int*)d_in[19];
  const int*   relidx   = (const int*)d_in[20];

  const float* h_last = hidden + (size_t)(L_ - 1) * B_ * T_ * H_;

  // workspace layout (floats), total ~12.06M floats ≈ 47 MB
  float* ws   = (float*)d_ws;
  float* cat    = ws;                                   // B*N*2H
  float* rel    = cat    + (size_t)B_ * N_ * 2 * H_;    // B*N*H
  float* u0buf  = rel    + (size_t)B_ * N_ * H_;        // B*N*H
  float* qbuf   = u0buf  + (size_t)B_ * N_ * H_;        // B*T*H
  float* kbuf   = qbuf   + (size_t)B_ * T_ * H_;        // B*N*H
  float* wdbuf  = kbuf   + (size_t)B_ * N_ * H_;        // B*T*H
  float* ugbuf  = wdbuf  + (size_t)B_ * T_ * H_;        // B*T*H
  float* scores = ugbuf  + (size_t)B_ * T_ * H_;        // B*T*N
  float* qtp    = scores + (size_t)B_ * T_ * N_;        // B*CH*T*H

  // 1) gathers
  k_gather<<<B_ * N_, H_, 0, stream>>>(se, rel_tab, heads, tails, relidx, cat, rel);

  // 2) u0 = (cat @ squeeze_w^T + b), masked    (M=4096, K=512)
  {
    dim3 g(B_ * N_ / 16, H_ / 16);
    k_gemm_nt<<<g, 32, 0, stream>>>(cat, squeeze_w, squeeze_b, tri_mask,
                                    u0buf, 2 * H_, 1.f);
  }
  // 3) q = (h_last @ wq^T + b) * H^-0.5        (M=2048, K=256)
  {
    dim3 g(B_ * T_ / 16, H_ / 16);
    k_gemm_nt<<<g, 32, 0, stream>>>(h_last, wq_w, wq_b, nullptr,
                                    qbuf, H_, 0.0625f);
    k_gemm_nt<<<g, 32, 0, stream>>>(h_last, wd_w, wd_b, nullptr,
                                    wdbuf, H_, 1.f);
    k_gemm_nt<<<g, 32, 0, stream>>>(h_last, wg_w, wg_b, nullptr,
                                    ugbuf, H_, 1.f);
  }
  // 4) k = rel @ wk^T + b                      (M=4096, K=256)
  {
    dim3 g(B_ * N_ / 16, H_ / 16);
    k_gemm_nt<<<g, 32, 0, stream>>>(rel, wk_w, wk_b, nullptr,
                                    kbuf, H_, 1.f);
  }
  // 5) scores + softmax
  {
    dim3 g(T_ / 16, N_ / 16, B_);
    k_scores<<<g, 32, 0, stream>>>(qbuf, kbuf, scores);
    k_softmax<<<B_ * T_, 256, 0, stream>>>(scores, tri_mask);
  }
  // 6) sequential scan (partials per n-chunk), then reduce
  k_scan<<<B_ * CH, 256, 0, stream>>>(u0buf, wdbuf, ugbuf, scores, h_last,
                                      aerfa_w, aerfa_b, wu_w, wu_b, qtp);
  k_reduce<<<(B_ * T_ * H_) / 256, 256, 0, stream>>>(qtp, (float*)d_out);
}